// CNSLoss_14104672600660
// MI455X (gfx1250) — compile-verified
//
#include <hip/hip_runtime.h>

// CDNA5 / gfx1250. wave32. Block = 256 threads = 8 waves.

typedef float v2f __attribute__((ext_vector_type(2)));
typedef float v8f __attribute__((ext_vector_type(8)));

#define RP1 97  // R + 1 columns per logits row

__device__ __forceinline__ float log_sigmoid(float x) {
    // log(sigmoid(x)) = min(x, 0) - log1p(exp(-|x|))   (numerically stable)
    return fminf(x, 0.0f) - log1pf(__expf(-fabsf(x)));
}

__global__ void cns_zero_out(float* __restrict__ out) {
    if (threadIdx.x == 0 && blockIdx.x == 0) out[0] = 0.0f;
}

__global__ void __launch_bounds__(256)
cns_loss_kernel(const float* __restrict__ logits,
                const int*   __restrict__ s_rows_b,
                const int*   __restrict__ s_cols_b,
                const int*   __restrict__ s_rows_h,
                const int*   __restrict__ s_cols_h,
                const float* __restrict__ s_base,
                const int*   __restrict__ m_t1_rows,
                const int*   __restrict__ m_t1_cols,
                const int*   __restrict__ m_t2_rows,
                const int*   __restrict__ m_t2_cols,
                const int*   __restrict__ m_head_cols,
                const float* __restrict__ m_base,
                float*       __restrict__ out,
                int n1, int n2)
{
    const int tid = threadIdx.x;
    const int i   = blockIdx.x * 256 + tid;

    float acc = 0.0f;
    if (i < n1) {
        // l1 = s_base + ls(logits[s_rows_b, s_cols_b]) - ls(logits[s_rows_h, s_cols_h])
        const int rb = s_rows_b[i], cb = s_cols_b[i];
        const int rh = s_rows_h[i], ch = s_cols_h[i];
        // Issue both scattered gathers before any VALU math so LOADcnt overlaps them.
        const float xb = logits[(size_t)rb * RP1 + cb];
        const float xh = logits[(size_t)rh * RP1 + ch];
        const float l1 = s_base[i] + log_sigmoid(xb) - log_sigmoid(xh);
        acc = fmaxf(l1, 0.0f);
    } else if (i < n1 + n2) {
        const int j  = i - n1;
        const int r1 = m_t1_rows[j], c1 = m_t1_cols[j];
        const int r2 = m_t2_rows[j], c2 = m_t2_cols[j];
        const int chd = m_head_cols[j];
        const float x1 = logits[(size_t)r1 * RP1 + c1];
        const float x2 = logits[(size_t)r2 * RP1 + c2];
        const float xh = logits[(size_t)r2 * RP1 + chd];
        const float l2 = m_base[j] + log_sigmoid(x1) + log_sigmoid(x2) - log_sigmoid(xh);
        acc = fmaxf(l2, 0.0f);
    }

    // ---- Block reduction of 256 per-lane partials via the f32 matrix pipe ----
    // Stage partials to LDS, then wave 0 folds them with 4 chained
    // V_WMMA_F32_16X16X4_F32 ops (B = ones => D[m][*] = row-sum, C-accumulated).
    __shared__ float sred[256];
    sred[tid] = acc;
    __syncthreads();

    if (tid < 32) {               // wave-uniform branch: EXEC all-ones in wave 0
        const int m  = tid & 15;            // A-matrix row for this lane
        const int k0 = (tid >> 4) << 1;     // lanes 0-15 -> K=0,1 ; lanes 16-31 -> K=2,3
        v2f b; b.x = 1.0f; b.y = 1.0f;      // B = ones(4x16)
        v8f c = {};
        #pragma unroll
        for (int chk = 0; chk < 4; ++chk) { // 4 x 64 partials
            v2f a;
            a.x = sred[chk * 64 + m * 4 + k0];
            a.y = sred[chk * 64 + m * 4 + k0 + 1];
            c = __builtin_amdgcn_wmma_f32_16x16x4_f32(
                    /*neg_a=*/false, a, /*neg_b=*/false, b,
                    /*c_mod=*/(short)0, c, /*reuse_a=*/false, /*reuse_b=*/false);
        }
        // Lane L holds D rows (L<16 ? 0..7 : 8..15) at column L%16; every column
        // carries identical row-sums, so:
        float s = c[0] + c[1] + c[2] + c[3] + c[4] + c[5] + c[6] + c[7];
        s += __shfl_xor(s, 16, 32);         // rows 0-7 + rows 8-15 = block total
        if (tid == 0) atomicAdd(out, s);    // global_atomic_add_f32, one per block
    }
}

extern "C" void kernel_launch(void* const* d_in, const int* in_sizes, int n_in,
                              void* d_out, int out_size, void* d_ws, size_t ws_size,
                              hipStream_t stream) {
    const float* logits     = (const float*)d_in[0];
    const int*   s_rows_b   = (const int*)  d_in[1];
    const int*   s_cols_b   = (const int*)  d_in[2];
    const int*   s_rows_h   = (const int*)  d_in[3];
    const int*   s_cols_h   = (const int*)  d_in[4];
    const float* s_base     = (const float*)d_in[5];
    const int*   m_t1_rows  = (const int*)  d_in[6];
    const int*   m_t1_cols  = (const int*)  d_in[7];
    const int*   m_t2_rows  = (const int*)  d_in[8];
    const int*   m_t2_cols  = (const int*)  d_in[9];
    const int*   m_head_cols= (const int*)  d_in[10];
    const float* m_base     = (const float*)d_in[11];
    float* out = (float*)d_out;

    const int n1 = in_sizes[1];                  // == len(s_rows_b) == len(s_base)
    int n2 = in_sizes[6];                        // elementwise bound for l2
    if (in_sizes[7]  < n2) n2 = in_sizes[7];
    if (in_sizes[8]  < n2) n2 = in_sizes[8];
    if (in_sizes[9]  < n2) n2 = in_sizes[9];
    if (in_sizes[10] < n2) n2 = in_sizes[10];
    if (in_sizes[11] < n2) n2 = in_sizes[11];

    const int total  = n1 + n2;
    const int blocks = (total + 255) / 256;

    cns_zero_out<<<1, 1, 0, stream>>>(out);
    cns_loss_kernel<<<blocks, 256, 0, stream>>>(
        logits, s_rows_b, s_cols_b, s_rows_h, s_cols_h, s_base,
        m_t1_rows, m_t1_cols, m_t2_rows, m_t2_cols, m_head_cols, m_base,
        out, n1, n2);
}